// LR_71674414235766
// MI455X (gfx1250) — compile-verified
//
#include <hip/hip_runtime.h>
#include <math.h>

typedef __attribute__((ext_vector_type(2))) float v2f;
typedef __attribute__((ext_vector_type(8))) float v8f;

#define BATCH   16384
#define NF      39
#define NMASK   13
#define EDIM    64
#define FDVOCAB 26000
#define BTILE   16          // batch rows per block (WMMA M)
#define NWAVES  4
#define TPB     (NWAVES * 32)

__global__ __launch_bounds__(TPB) void lr_emb_wmma_kernel(
    const int*   __restrict__ features,    // [B, NF]
    const int*   __restrict__ mask,        // [B, NMASK]
    const int*   __restrict__ mask_apply,  // [B, NMASK] (0/1)
    const float* __restrict__ mask_value,  // [B, NMASK]
    const float* __restrict__ emb,         // [NF*FDVOCAB, EDIM]
    const float* __restrict__ lin_w,       // [EDIM]
    const float* __restrict__ lin_b,       // scalar
    float*       __restrict__ out)         // [B]
{
    __shared__ float s_lds[BTILE * NF];        // per-(b,f) multiplicative scale
    __shared__ float p_lds[NWAVES * BTILE];    // per-wave partial logits

    const int tid     = threadIdx.x;
    const int lane    = tid & 31;              // wave32
    const int wave    = tid >> 5;
    const int b0      = blockIdx.x * BTILE;
    const int b_local = lane & 15;             // A-matrix row M
    const int koff    = (lane >> 4) << 1;      // K slot base: lanes 0-15 -> K{0,1}, 16-31 -> K{2,3}

    // ---- per-(b,f) scale = product over mask hits (handles duplicate indices) ----
    for (int idx = tid; idx < BTILE * NF; idx += TPB) {
        int bl = idx / NF;
        int f  = idx - bl * NF;
        int bg = b0 + bl;
        float s = 1.0f;
        #pragma unroll
        for (int j = 0; j < NMASK; ++j) {
            if (mask[bg * NMASK + j] == f) {
                float mv = mask_value[bg * NMASK + j];
                s *= (mask_apply[bg * NMASK + j] != 0) ? mv : 1.0f;
            }
        }
        s_lds[idx] = s;
    }

    // ---- loop-invariant B-matrix fragments of lin_w: wv[c] = w[4c+koff .. 4c+koff+1] ----
    v2f wv[EDIM / 4];
    #pragma unroll
    for (int c = 0; c < EDIM / 4; ++c)
        wv[c] = *(const v2f*)(lin_w + 4 * c + koff);

    __syncthreads();

    // ---- gather + WMMA f32 16x16x4 accumulation over (f, e) ----
    v8f acc0 = {};
    v8f acc1 = {};
    for (int f = wave; f < NF; f += NWAVES) {   // wave-uniform: EXEC stays all-ones
        int   fidx = features[(b0 + b_local) * NF + f];
        float s    = s_lds[b_local * NF + f];
        const float* row = emb + ((size_t)fidx + (size_t)f * FDVOCAB) * EDIM + koff;
        #pragma unroll
        for (int c = 0; c < EDIM / 4; c += 2) {
            v2f a0 = *(const v2f*)(row + 4 * c);        // 8B aligned: row is 256B-aligned, koff even
            v2f a1 = *(const v2f*)(row + 4 * (c + 1));
            a0 *= s;
            a1 *= s;
            acc0 = __builtin_amdgcn_wmma_f32_16x16x4_f32(
                       false, a0, false, wv[c],     (short)0, acc0, false, false);
            acc1 = __builtin_amdgcn_wmma_f32_16x16x4_f32(
                       false, a1, false, wv[c + 1], (short)0, acc1, false, false);
        }
    }

    // ---- extract column N=0 of D: lane 0 holds M=0..7 in VGPRs 0..7, lane 16 holds M=8..15 ----
    if (lane == 0) {
        #pragma unroll
        for (int r = 0; r < 8; ++r) p_lds[wave * BTILE + r] = acc0[r] + acc1[r];
    } else if (lane == 16) {
        #pragma unroll
        for (int r = 0; r < 8; ++r) p_lds[wave * BTILE + 8 + r] = acc0[r] + acc1[r];
    }
    __syncthreads();

    // ---- cross-wave reduce, bias, sigmoid ----
    if (tid < BTILE) {
        float logit = 0.0f;
        #pragma unroll
        for (int w = 0; w < NWAVES; ++w) logit += p_lds[w * BTILE + tid];
        logit += (float)NF * lin_b[0];
        out[b0 + tid] = 1.0f / (1.0f + expf(-logit));
    }
}

extern "C" void kernel_launch(void* const* d_in, const int* in_sizes, int n_in,
                              void* d_out, int out_size, void* d_ws, size_t ws_size,
                              hipStream_t stream) {
    (void)in_sizes; (void)n_in; (void)d_ws; (void)ws_size; (void)out_size;
    const int*   features   = (const int*)d_in[0];
    const int*   mask       = (const int*)d_in[1];
    const int*   mask_apply = (const int*)d_in[2];
    const float* mask_value = (const float*)d_in[3];
    const float* emb        = (const float*)d_in[4];
    const float* lin_w      = (const float*)d_in[5];
    const float* lin_b      = (const float*)d_in[6];
    float*       out        = (float*)d_out;

    dim3 grid(BATCH / BTILE);
    dim3 block(TPB);
    lr_emb_wmma_kernel<<<grid, block, 0, stream>>>(
        features, mask, mask_apply, mask_value, emb, lin_w, lin_b, out);
}